// CodeSwitchLoss_13494787244307
// MI455X (gfx1250) — compile-verified
//
#include <hip/hip_runtime.h>
#include <math.h>

typedef __attribute__((ext_vector_type(2))) float v2f;
typedef __attribute__((ext_vector_type(8))) float v8f;

#define DDIM  1024
#define NROWS 4096
#define INVT  10.0f      // 1 / TEMPERATURE
#define LDS_STRIDE 36    // KC + pad: 36*4=144 bytes -> 16B-aligned rows, conflict-free
#define BM 128
#define BN 64
#define KC 32
#define NCHUNK (DDIM / KC)

// ptrtoint of an LDS (addrspace 3) pointer == wave-relative LDS byte offset,
// which is what GLOBAL_LOAD_ASYNC_TO_LDS expects in its VDST VGPR.
__device__ __forceinline__ unsigned lds_off(const float* p) {
  return (unsigned)(size_t)(__attribute__((address_space(3))) const float*)p;
}

// ---------------------------------------------------------------------------
// Kernel A: row-wise L2 normalization of the 4 input matrices into X[4096][1024]
// Row ordering matches reference V = stack([e, k, etk, kte]).
// ---------------------------------------------------------------------------
__global__ __launch_bounds__(256) void normalize_rows(
    const float* __restrict__ english, const float* __restrict__ etok,
    const float* __restrict__ ktoe,    const float* __restrict__ korean,
    float* __restrict__ X) {
  const int row = blockIdx.x;
  const int a = row >> 10;
  const int i = row & 1023;
  const float* src;
  if      (a == 0) src = english + (size_t)i * DDIM;
  else if (a == 1) src = korean  + (size_t)i * DDIM;
  else if (a == 2) src = etok    + (size_t)i * DDIM;
  else             src = ktoe    + (size_t)i * DDIM;

  const int tid = threadIdx.x;
  float4 v = *(const float4*)&src[tid * 4];
  float ss = v.x * v.x + v.y * v.y + v.z * v.z + v.w * v.w;

  __shared__ float buf[256];
  buf[tid] = ss;
  __syncthreads();
  for (int s = 128; s > 0; s >>= 1) {
    if (tid < s) buf[tid] += buf[tid + s];
    __syncthreads();
  }
  const float inv = 1.0f / sqrtf(buf[0]);
  float4 o;
  o.x = v.x * inv; o.y = v.y * inv; o.z = v.z * inv; o.w = v.w * inv;
  *(float4*)&X[(size_t)row * DDIM + tid * 4] = o;
}

// ---------------------------------------------------------------------------
// Kernel Z: zero the row-sum accumulator (required each launch; atomics follow)
// ---------------------------------------------------------------------------
__global__ __launch_bounds__(256) void zero_rowsum(float* __restrict__ p) {
  p[blockIdx.x * 256 + threadIdx.x] = 0.0f;
}

// ---------------------------------------------------------------------------
// Kernel B: per-index i, 4x4 exp-gram -> pos_sum/self, plus the two
// code-switch regularizer norms. One block (256 threads) per i.
// ---------------------------------------------------------------------------
__global__ __launch_bounds__(256) void pos_reg_kernel(
    const float* __restrict__ X, const float* __restrict__ ratios,
    float* __restrict__ possum, float* __restrict__ selfe,
    float* __restrict__ regv) {
  const int i = blockIdx.x;
  const int tid = threadIdx.x;
  const float r = ratios[i];
  const float omr = 1.0f - r;

  const float* e   = X + ((size_t)0 * 1024 + i) * DDIM;
  const float* kk  = X + ((size_t)1 * 1024 + i) * DDIM;
  const float* etk = X + ((size_t)2 * 1024 + i) * DDIM;
  const float* kte = X + ((size_t)3 * 1024 + i) * DDIM;

  float4 E4 = *(const float4*)&e[tid * 4];
  float4 K4 = *(const float4*)&kk[tid * 4];
  float4 T4 = *(const float4*)&etk[tid * 4];
  float4 U4 = *(const float4*)&kte[tid * 4];
  const float* Ea = (const float*)&E4;
  const float* Ka = (const float*)&K4;
  const float* Ta = (const float*)&T4;
  const float* Ua = (const float*)&U4;

  float p[12];
#pragma unroll
  for (int q = 0; q < 12; ++q) p[q] = 0.0f;
#pragma unroll
  for (int q = 0; q < 4; ++q) {
    const float a = Ea[q], b = Ka[q], c = Ta[q], d = Ua[q];
    p[0] += a * a; p[1] += a * b; p[2] += a * c; p[3] += a * d;
    p[4] += b * b; p[5] += b * c; p[6] += b * d;
    p[7] += c * c; p[8] += c * d; p[9] += d * d;
    const float m1 = c - (r * a + omr * b);
    const float m2 = d - (omr * a + r * b);
    p[10] += m1 * m1;
    p[11] += m2 * m2;
  }

  __shared__ float buf[256];
  __shared__ float res[12];
  for (int q = 0; q < 12; ++q) {
    buf[tid] = p[q];
    __syncthreads();
    for (int s = 128; s > 0; s >>= 1) {
      if (tid < s) buf[tid] += buf[tid + s];
      __syncthreads();
    }
    if (tid == 0) res[q] = buf[0];
    __syncthreads();
  }

  if (tid == 0) {
    float d[4][4];
    d[0][0] = res[0];
    d[0][1] = d[1][0] = res[1];
    d[0][2] = d[2][0] = res[2];
    d[0][3] = d[3][0] = res[3];
    d[1][1] = res[4];
    d[1][2] = d[2][1] = res[5];
    d[1][3] = d[3][1] = res[6];
    d[2][2] = res[7];
    d[2][3] = d[3][2] = res[8];
    d[3][3] = res[9];
#pragma unroll
    for (int a = 0; a < 4; ++a) {
      float same = 0.0f;
#pragma unroll
      for (int b = 0; b < 4; ++b) same += expf(d[a][b] * INVT);
      const float sf = expf(d[a][a] * INVT);
      possum[a * 1024 + i] = same - sf;
      selfe[a * 1024 + i]  = sf;
    }
    regv[i] = sqrtf(res[10]) + sqrtf(res[11]);
  }
}

// ---------------------------------------------------------------------------
// Kernel C: Gram + exp + row sums via V_WMMA_F32_16X16X4_F32.
// Block tile 128x64; 8 waves (4x2); each wave owns a 2x2 grid of 16x16 tiles.
// K staged in 32-wide chunks through double-buffered LDS filled by
// GLOBAL_LOAD_ASYNC_TO_LDS_B128 (ASYNCcnt), overlapping DMA with WMMA.
// ---------------------------------------------------------------------------
__global__ __launch_bounds__(256) void gram_exp_rowsum(
    const float* __restrict__ X, float* __restrict__ rowsum) {
  __shared__ float sA[2][BM * LDS_STRIDE];
  __shared__ float sB[2][BN * LDS_STRIDE];

  const int r0 = blockIdx.y * BM;
  const int c0 = blockIdx.x * BN;
  const int tid  = threadIdx.x;
  const int lane = tid & 31;
  const int wave = tid >> 5;
  const int wr = wave >> 1;          // 0..3 : 32-row strip
  const int wc = wave & 1;           // 0..1 : 32-col strip
  const int m    = lane & 15;
  const int koff = (lane >> 4) << 1; // lanes 16-31 -> K+2

  // Per-thread staging addresses. A: 4 x b128, B: 2 x b128 per chunk.
  unsigned long long gA[4], gB[2];
  unsigned lA[2][4], lB[2][2];
#pragma unroll
  for (int j = 0; j < 4; ++j) {
    const int idx = tid + j * 256;
    const int rr  = idx >> 3;
    const int c4  = (idx & 7) << 2;
    gA[j] = (unsigned long long)(X + (size_t)(r0 + rr) * DDIM + c4);
    lA[0][j] = lds_off(&sA[0][rr * LDS_STRIDE + c4]);
    lA[1][j] = lds_off(&sA[1][rr * LDS_STRIDE + c4]);
  }
#pragma unroll
  for (int j = 0; j < 2; ++j) {
    const int idx = tid + j * 256;
    const int rr  = idx >> 3;
    const int c4  = (idx & 7) << 2;
    gB[j] = (unsigned long long)(X + (size_t)(c0 + rr) * DDIM + c4);
    lB[0][j] = lds_off(&sB[0][rr * LDS_STRIDE + c4]);
    lB[1][j] = lds_off(&sB[1][rr * LDS_STRIDE + c4]);
  }

  // Issue async DMA for one K-chunk into buffer `buf`, then advance gptrs.
  auto stage = [&](int buf) {
#pragma unroll
    for (int j = 0; j < 4; ++j) {
      asm volatile("global_load_async_to_lds_b128 %0, %1, off"
                   :: "v"(lA[buf][j]), "v"(gA[j]) : "memory");
      gA[j] += KC * sizeof(float);
    }
#pragma unroll
    for (int j = 0; j < 2; ++j) {
      asm volatile("global_load_async_to_lds_b128 %0, %1, off"
                   :: "v"(lB[buf][j]), "v"(gB[j]) : "memory");
      gB[j] += KC * sizeof(float);
    }
  };

  v8f acc00 = {}, acc01 = {}, acc10 = {}, acc11 = {};

  stage(0);
  asm volatile("s_wait_asynccnt 0x0" ::: "memory");
  __syncthreads();

  for (int chunk = 0; chunk < NCHUNK; ++chunk) {
    const int cur = chunk & 1;
    if (chunk + 1 < NCHUNK) stage(cur ^ 1);  // DMA next chunk while computing

    const float* pa = &sA[cur][(wr * 32 + m) * LDS_STRIDE + koff];
    const float* pb = &sB[cur][(wc * 32 + m) * LDS_STRIDE + koff];
#pragma unroll
    for (int k4 = 0; k4 < KC; k4 += 4) {
      v2f a0 = *(const v2f*)(pa + k4);
      v2f a1 = *(const v2f*)(pa + 16 * LDS_STRIDE + k4);
      v2f b0 = *(const v2f*)(pb + k4);
      v2f b1 = *(const v2f*)(pb + 16 * LDS_STRIDE + k4);
      acc00 = __builtin_amdgcn_wmma_f32_16x16x4_f32(false, a0, false, b0, (short)0, acc00, false, false);
      acc01 = __builtin_amdgcn_wmma_f32_16x16x4_f32(false, a0, false, b1, (short)0, acc01, false, false);
      acc10 = __builtin_amdgcn_wmma_f32_16x16x4_f32(false, a1, false, b0, (short)0, acc10, false, false);
      acc11 = __builtin_amdgcn_wmma_f32_16x16x4_f32(false, a1, false, b1, (short)0, acc11, false, false);
    }

    asm volatile("s_wait_asynccnt 0x0" ::: "memory");
    __syncthreads();
  }

  // exp(S/t); the two col-tiles of a row-strip share rows -> combine before
  // the width-16 shuffle reduction. C layout: VGPR p -> M=p (lanes 0-15),
  // M=p+8 (lanes 16-31), N = lane&15.
#pragma unroll
  for (int p = 0; p < 8; ++p) {
    float v0 = expf(acc00[p] * INVT) + expf(acc01[p] * INVT);
    v0 += __shfl_xor(v0, 1, 16);
    v0 += __shfl_xor(v0, 2, 16);
    v0 += __shfl_xor(v0, 4, 16);
    v0 += __shfl_xor(v0, 8, 16);
    float v1 = expf(acc10[p] * INVT) + expf(acc11[p] * INVT);
    v1 += __shfl_xor(v1, 1, 16);
    v1 += __shfl_xor(v1, 2, 16);
    v1 += __shfl_xor(v1, 4, 16);
    v1 += __shfl_xor(v1, 8, 16);
    if ((lane & 15) == 0) {
      const int rbase = r0 + wr * 32 + p + ((lane >> 4) << 3);
      atomicAdd(&rowsum[rbase], v0);
      atomicAdd(&rowsum[rbase + 16], v1);
    }
  }
}

// ---------------------------------------------------------------------------
// Kernel D: final scalar reduction.
// contrastive[a,i] = -log(pos / (pos + neg)) = log((rowtotal - self) / pos)
// ---------------------------------------------------------------------------
__global__ __launch_bounds__(256) void finalize(
    const float* __restrict__ rowsum, const float* __restrict__ possum,
    const float* __restrict__ selfe,  const float* __restrict__ regv,
    float* __restrict__ out) {
  const int tid = threadIdx.x;
  float cs = 0.0f;
  for (int r = tid; r < NROWS; r += 256)
    cs += logf((rowsum[r] - selfe[r]) / possum[r]);
  float rg = 0.0f;
  for (int i = tid; i < 1024; i += 256) rg += regv[i];

  __shared__ float buf[256];
  buf[tid] = cs;
  __syncthreads();
  for (int s = 128; s > 0; s >>= 1) {
    if (tid < s) buf[tid] += buf[tid + s];
    __syncthreads();
  }
  const float csum = buf[0];
  __syncthreads();
  buf[tid] = rg;
  __syncthreads();
  for (int s = 128; s > 0; s >>= 1) {
    if (tid < s) buf[tid] += buf[tid + s];
    __syncthreads();
  }
  if (tid == 0) out[0] = (csum + 0.5f * buf[0]) / 1024.0f;
}

// ---------------------------------------------------------------------------
extern "C" void kernel_launch(void* const* d_in, const int* in_sizes, int n_in,
                              void* d_out, int out_size, void* d_ws, size_t ws_size,
                              hipStream_t stream) {
  (void)in_sizes; (void)n_in; (void)out_size; (void)ws_size;
  const float* english = (const float*)d_in[0];
  const float* etok    = (const float*)d_in[1];
  const float* ktoe    = (const float*)d_in[2];
  const float* korean  = (const float*)d_in[3];
  const float* ratios  = (const float*)d_in[4];

  float* ws     = (float*)d_ws;
  float* X      = ws;                        // 4096*1024 floats (16 MB)
  float* rowsum = X + (size_t)NROWS * DDIM;  // 4096
  float* possum = rowsum + NROWS;            // 4096
  float* selfe  = possum + NROWS;            // 4096
  float* regv   = selfe + NROWS;             // 1024

  normalize_rows<<<NROWS, 256, 0, stream>>>(english, etok, ktoe, korean, X);
  zero_rowsum<<<NROWS / 256, 256, 0, stream>>>(rowsum);
  pos_reg_kernel<<<1024, 256, 0, stream>>>(X, ratios, possum, selfe, regv);
  dim3 grid(NROWS / BN, NROWS / BM);  // (col tiles, row tiles)
  gram_exp_rowsum<<<grid, 256, 0, stream>>>(X, rowsum);
  finalize<<<1, 256, 0, stream>>>(rowsum, possum, selfe, regv, (float*)d_out);
}